// MPA2_37056977830475
// MI455X (gfx1250) — compile-verified
//
#include <hip/hip_runtime.h>
#include <hip/hip_bf16.h>

// Q[m,n] = IVF[m, idx[0][n], n] * IVF[m, idx[1][n], n] * wout[m,n] * (1/num_M)
//
// num_M=4, num_VN=6, K=3 -> 24 outputs, 72 FLOPs, <1 KB traffic.
// Pure launch-latency bound on MI455X (23.3 TB/s HBM). Optimal shape:
// one wave32, one lane per output, dependent gather chain, minimal
// instruction count. De-linearization is done by launching a 2D block
// (num_VN, num_M) so m/n come from hardware thread IDs instead of a
// ~25-instruction runtime integer division.
//
// WMMA intentionally absent: elementwise gather-product has no contraction
// dimension for the 16x16xK matrix pipes, and the whole problem is smaller
// than one WMMA accumulator tile; a v_wmma would only add XDL hazard latency
// to a latency-bound wave.
__global__ __launch_bounds__(32) void MPA2_37056977830475_kernel(
    const float* __restrict__ IVF,      // [num_M, K, num_VN] row-major
    const int*   __restrict__ VN_index, // [2, num_VN]
    const float* __restrict__ wout,     // [num_M, num_VN]
    float*       __restrict__ Q,        // [num_M, num_VN]
    int num_VN, int K, float inv_M)
{
    const int n = threadIdx.x;          // 0..num_VN-1  (hardware-provided)
    const int m = threadIdx.y;          // 0..num_M-1   (hardware-provided)
    const int t = m * num_VN + n;

    // Gather indices (int32 per harness convention for integer inputs).
    const int i0 = VN_index[n];
    const int i1 = VN_index[num_VN + n];

    // Two gathered reads from IVF; issue back-to-back, one s_wait_loadcnt.
    const float a = IVF[(m * K + i0) * num_VN + n];
    const float b = IVF[(m * K + i1) * num_VN + n];
    const float w = wout[t];

    Q[t] = a * b * (w * inv_M);
}

extern "C" void kernel_launch(void* const* d_in, const int* in_sizes, int n_in,
                              void* d_out, int out_size, void* d_ws, size_t ws_size,
                              hipStream_t stream) {
    // Input order (from setup_inputs): num_M, num_VN, IVF, VN_index, wout.
    // Shape parameters recovered on the host from size metadata, so the
    // device never reads the scalar inputs (shorter critical path).
    const float* IVF      = (const float*)d_in[2];
    const int*   VN_index = (const int*)  d_in[3];
    const float* wout     = (const float*)d_in[4];
    float*       Q        = (float*)d_out;

    const int num_VN = in_sizes[3] / 2;                    // VN_index is [2, num_VN]
    const int num_M  = (num_VN > 0) ? (out_size / num_VN) : 1;
    const int K      = (num_M * num_VN > 0) ? (in_sizes[2] / (num_M * num_VN)) : 1;
    const float inv_M = 1.0f / (float)num_M;

    // One 2D block (num_VN x num_M) = 24 threads = a single wave32.
    dim3 block((unsigned)num_VN, (unsigned)num_M, 1);
    MPA2_37056977830475_kernel<<<1, block, 0, stream>>>(
        IVF, VN_index, wout, Q, num_VN, K, inv_M);
}